// RadianceField_26345329394043
// MI455X (gfx1250) — compile-verified
//
#include <hip/hip_runtime.h>
#include <stdint.h>

#define IDIM 160
#define GDIM 161                       // grid nodes per axis (IDIM+1)
#define NB_SAMPLES 64
#define NB_RAYS 32768
#define NCELL (GDIM * GDIM * GDIM)     // 4,173,281
#define FUSE_CELLS_PER_BLOCK 256
#define RENDER_BLOCK 128

typedef __attribute__((ext_vector_type(4))) unsigned int u32x4;
typedef __attribute__((ext_vector_type(4))) int          i32x4;
typedef __attribute__((ext_vector_type(8))) int          i32x8;
typedef __attribute__((ext_vector_type(2))) float        v2f;
typedef __attribute__((ext_vector_type(8))) float        v8f;

// ---------------------------------------------------------------------------
// LDS byte offset of a __shared__ object: generic LDS pointers on gfx1250 are
// {SHARED_BASE[63:32], lds_offset[31:0]}, so truncation yields the offset.
// ---------------------------------------------------------------------------
__device__ __forceinline__ unsigned lds_offset_of(const void* p) {
  return (unsigned)(unsigned long long)(uintptr_t)p;
}

// ---------------------------------------------------------------------------
// Tensor Data Mover: 1-row (1D) contiguous copy of `ndw` dwords from global
// memory to LDS.  D# per CDNA5 ISA:
//   group0: count=1 | lds_addr | global_addr[56:0] | type=2
//   group1: data_size=4B, tensor_dim0 = tile_dim0 = ndw, tensor_dim1 = 1,
//           tensor_dim0_stride = ndw, no padding / iterate / multicast
//   groups 2,3 zero (<=2D tensor).  Tracked with TENSORcnt.
// ---------------------------------------------------------------------------
__device__ __forceinline__ void tdm_load_1d_to_lds(const void* gptr,
                                                   unsigned lds_addr,
                                                   unsigned ndw) {
  unsigned long long ga = (unsigned long long)(uintptr_t)gptr;
  u32x4 g0;
  g0[0] = 1u;                                            // count=1, user mode
  g0[1] = lds_addr;                                      // lds_addr (bytes)
  g0[2] = (unsigned)(ga & 0xffffffffu);                  // global_addr[31:0]
  g0[3] = (unsigned)((ga >> 32) & 0x01ffffffu)           // global_addr[56:32]
        | (2u << 30);                                    // type = 2 ("image")

  i32x8 g1;
  g1[0] = (int)(2u << 16);                               // data_size = 4B
  g1[1] = (int)((ndw & 0xffffu) << 16);                  // tensor_dim0[15:0]
  g1[2] = (int)(((ndw >> 16) & 0xffffu)                  // tensor_dim0[31:16]
        | (1u << 16));                                   // tensor_dim1 = 1 (lo)
  g1[3] = (int)((ndw & 0xffffu) << 16);                  // tile_dim0 = ndw
  g1[4] = 0;                                             // tile_dim1 = tile_dim2 = 0
  g1[5] = (int)ndw;                                      // tensor_dim0_stride[31:0]
  g1[6] = 0;
  g1[7] = 0;

  i32x4 z4 = {0, 0, 0, 0};
#if __clang_major__ >= 23
  i32x8 z8 = {0, 0, 0, 0, 0, 0, 0, 0};
  __builtin_amdgcn_tensor_load_to_lds(g0, g1, z4, z4, z8, 0);
#else
  __builtin_amdgcn_tensor_load_to_lds(g0, g1, z4, z4, 0);
#endif
}

// ---------------------------------------------------------------------------
// Kernel 1: fuse {sum of 9 SH channels, opacity} into one float2 table.
//
// TDM -> LDS -> WMMA pipeline:
//  * wave 0 DMAs the block's contiguous 256-cell (9216 B) grid chunk to LDS,
//    waits TENSORcnt, workgroup barrier.
//  * each wave reduces 16 cells at a time with V_WMMA_F32_16X16X4_F32:
//      A = all-ones 16x4, B[k][n] = channel k of cell n  =>
//      D[m][n] = sum_k B[k][n]  (channel sum of cell n, replicated over m).
//    Two accumulating K=4 WMMAs cover channels 0..7; channel 8 added scalar.
//    Lane n (n<16) reads its cell's sum directly from c[0] -- no shuffles.
//  * EXEC is all-ones at the WMMA (store masking happens afterwards).
// ---------------------------------------------------------------------------
__global__ void __launch_bounds__(FUSE_CELLS_PER_BLOCK)
fuse_kernel(const float* __restrict__ grid, const float* __restrict__ opacity,
            float2* __restrict__ fused, int ncell) {
  __shared__ float lgrid[FUSE_CELLS_PER_BLOCK * 9];

  int base  = blockIdx.x * FUSE_CELLS_PER_BLOCK;
  int cells = ncell - base;
  if (cells > FUSE_CELLS_PER_BLOCK) cells = FUSE_CELLS_PER_BLOCK;
  unsigned ndw = (unsigned)cells * 9u;

  if ((threadIdx.x >> 5) == 0) {   // wave 0 issues the DMA (EXEC ignored by TDM)
    tdm_load_1d_to_lds(grid + (size_t)base * 9u, lds_offset_of(lgrid), ndw);
    __builtin_amdgcn_s_wait_tensorcnt(0);
  }
  __syncthreads();

  const int lane = (int)threadIdx.x & 31;
  const int wave = (int)threadIdx.x >> 5;
  const int n    = lane & 15;      // column (cell within 16-cell tile)
  const int kh   = lane >> 4;      // K-half (channels {0,1} vs {2,3})

  v2f ones;
  ones[0] = 1.0f;
  ones[1] = 1.0f;

#pragma unroll
  for (int it = 0; it < 2; ++it) {                 // 8 waves * 2 tiles * 16 = 256 cells
    const int cellT = (wave * 2 + it) * 16;
    const int cell  = cellT + n;                   // always < 256: in-bounds LDS
    const int cb    = cell * 9;

    v2f b0, b1;
    b0[0] = lgrid[cb + 2 * kh];                    // channels 0..3
    b0[1] = lgrid[cb + 2 * kh + 1];
    b1[0] = lgrid[cb + 4 + 2 * kh];                // channels 4..7
    b1[1] = lgrid[cb + 4 + 2 * kh + 1];

    v8f c = {};
    c = __builtin_amdgcn_wmma_f32_16x16x4_f32(false, ones, false, b0,
                                              (short)0, c, false, false);
    c = __builtin_amdgcn_wmma_f32_16x16x4_f32(false, ones, false, b1,
                                              (short)0, c, false, false);

    if (kh == 0 && cell < cells) {                 // lanes 0..15 own columns 0..15
      float2 o;
      o.x = c[0] + lgrid[cb + 8];                  // + channel 8
      o.y = opacity[base + cell];
      fused[base + cell] = o;
    }
  }
}

// ---------------------------------------------------------------------------
// Deterministic hash -> U[0,1)  (stand-in for jax threefry; deterministic
// per (ray, sample) so replays are bit-identical).
// ---------------------------------------------------------------------------
__device__ __forceinline__ float hash_u01(unsigned s) {
  s ^= s >> 17; s *= 0xed5ad4bbu;
  s ^= s >> 11; s *= 0xac4c1b51u;
  s ^= s >> 15; s *= 0x31848babu;
  s ^= s >> 14;
  return (float)(s >> 8) * (1.0f / 16777216.0f);
}

// ---------------------------------------------------------------------------
// Kernel 2: one thread per ray (wave32: 32 rays/wave).
//  - slab AABB test (matches reference clamping)
//  - 64 sample t's, register-resident fully-unrolled bitonic sort
//  - global_prefetch of sample-0 corner lines before the sort hides latency
//  - sorted t staged to LDS, then a dynamic 63-iter gather loop over the
//    L2-resident fused float2 table (one global_load_b64 per corner)
//  - streaming transmittance accumulation (no materialized cumsum)
// ---------------------------------------------------------------------------
__global__ void __launch_bounds__(RENDER_BLOCK)
render_kernel(const float* __restrict__ xin, const float* __restrict__ din,
              const float2* __restrict__ fused, float* __restrict__ out) {
  __shared__ float ts[NB_SAMPLES][RENDER_BLOCK];

  const int tid = (int)threadIdx.x;
  const int ray = (int)blockIdx.x * RENDER_BLOCK + tid;

  const float px = xin[ray * 3 + 0], py = xin[ray * 3 + 1], pz = xin[ray * 3 + 2];
  const float dx = din[ray * 3 + 0], dy = din[ray * 3 + 1], dz = din[ray * 3 + 2];

  const float inf = (float)IDIM * (float)IDIM * (float)IDIM;
  const float ivx = 1.0f / dx, ivy = 1.0f / dy, ivz = 1.0f / dz;

  const float t0x = (0.0f - px) * ivx, t1x = ((float)IDIM - px) * ivx;
  const float t0y = (0.0f - py) * ivy, t1y = ((float)IDIM - py) * ivy;
  const float t0z = (0.0f - pz) * ivz, t1z = ((float)IDIM - pz) * ivz;

  const float tmin = fmaxf(-inf, fmaxf(fminf(t0x, t1x),
                           fmaxf(fminf(t0y, t1y), fminf(t0z, t1z))));
  const float tmax = fminf(inf, fminf(fmaxf(t0x, t1x),
                           fminf(fmaxf(t0y, t1y), fmaxf(t0z, t1z))));
  const float range = tmax - tmin;

  // --- generate 64 sample positions, track the minimum for prefetch ---
  float t[NB_SAMPLES];
  float tfirst = inf;
#pragma unroll
  for (int i = 0; i < NB_SAMPLES; ++i) {
    float u = hash_u01(((unsigned)ray << 6) + (unsigned)i + 0x9e3779b9u);
    t[i] = fmaf(u, range, tmin);
    tfirst = fminf(tfirst, t[i]);
  }

  // --- prefetch sample 0's corner lines; sort hides the latency ---
  {
    float q0 = fmaf(tfirst, dx, px);
    float q1 = fmaf(tfirst, dy, py);
    float q2 = fmaf(tfirst, dz, pz);
    int ix = (int)fminf(fmaxf(floorf(q0), 0.0f), (float)(IDIM - 1));
    int iy = (int)fminf(fmaxf(floorf(q1), 0.0f), (float)(IDIM - 1));
    int iz = (int)fminf(fmaxf(floorf(q2), 0.0f), (float)(IDIM - 1));
    const float2* c = fused + ((ix * GDIM + iy) * GDIM + iz);
    __builtin_prefetch(c, 0, 3);                 // x0 face -> global_prefetch
    __builtin_prefetch(c + GDIM * GDIM, 0, 3);   // x1 face
  }

  // --- fully-unrolled bitonic sort (compile-time indices => registers) ---
#pragma unroll
  for (int k = 2; k <= NB_SAMPLES; k <<= 1) {
#pragma unroll
    for (int j = k >> 1; j > 0; j >>= 1) {
#pragma unroll
      for (int i = 0; i < NB_SAMPLES; ++i) {
        const int l = i ^ j;
        if (l > i) {
          const bool up = ((i & k) == 0);
          float a = t[i], b = t[l];
          bool sw = up ? (a > b) : (a < b);
          float lo = sw ? b : a;
          float hi = sw ? a : b;
          t[i] = lo;
          t[l] = hi;
        }
      }
    }
  }

  // --- stage sorted t's to LDS so the gather loop can index dynamically ---
#pragma unroll
  for (int i = 0; i < NB_SAMPLES; ++i) ts[i][tid] = t[i];

  // --- march: gather fused {h_sum, opacity}, accumulate radiance ---
  float T = 1.0f;
  float acc = 0.0f;
  float tc = ts[0][tid];

  for (int s = 0; s < NB_SAMPLES - 1; ++s) {
    float tn = ts[s + 1][tid];
    float delta = tn - tc;

    float q0 = fmaf(tc, dx, px);
    float q1 = fmaf(tc, dy, py);
    float q2 = fmaf(tc, dz, pz);

    float f0 = fminf(fmaxf(floorf(q0), 0.0f), (float)(IDIM - 1));
    float f1 = fminf(fmaxf(floorf(q1), 0.0f), (float)(IDIM - 1));
    float f2 = fminf(fmaxf(floorf(q2), 0.0f), (float)(IDIM - 1));
    float fx = q0 - f0, fy = q1 - f1, fz = q2 - f2;
    int ix = (int)f0, iy = (int)f1, iz = (int)f2;

    const float2* c = fused + ((ix * GDIM + iy) * GDIM + iz);
    float2 v000 = c[0];
    float2 v001 = c[1];
    float2 v010 = c[GDIM];
    float2 v011 = c[GDIM + 1];
    float2 v100 = c[GDIM * GDIM];
    float2 v101 = c[GDIM * GDIM + 1];
    float2 v110 = c[GDIM * GDIM + GDIM];
    float2 v111 = c[GDIM * GDIM + GDIM + 1];

    float gx = 1.0f - fx, gy = 1.0f - fy, gz = 1.0f - fz;
    float w00 = gx * gy, w01 = gx * fy, w10 = fx * gy, w11 = fx * fy;
    float w000 = w00 * gz, w001 = w00 * fz;
    float w010 = w01 * gz, w011 = w01 * fz;
    float w100 = w10 * gz, w101 = w10 * fz;
    float w110 = w11 * gz, w111 = w11 * fz;

    float h = w000 * v000.x + w001 * v001.x + w010 * v010.x + w011 * v011.x +
              w100 * v100.x + w101 * v101.x + w110 * v110.x + w111 * v111.x;
    float o = w000 * v000.y + w001 * v001.y + w010 * v010.y + w011 * v011.y +
              w100 * v100.y + w101 * v101.y + w110 * v110.y + w111 * v111.y;

    float cur = delta * o;
    float e   = __expf(-cur);                 // exp(-delta*sigma)
    float sig = 1.0f / (1.0f + __expf(-h));   // sigmoid(sum of channels)

    acc = fmaf(T * (1.0f - e), sig, acc);
    T *= e;
    tc = tn;
  }

  out[ray] = acc;
}

// ---------------------------------------------------------------------------
extern "C" void kernel_launch(void* const* d_in, const int* in_sizes, int n_in,
                              void* d_out, int out_size, void* d_ws, size_t ws_size,
                              hipStream_t stream) {
  const float* x       = (const float*)d_in[0];   // [NB_RAYS,3]
  const float* d       = (const float*)d_in[1];   // [NB_RAYS,3]
  const float* grid    = (const float*)d_in[2];   // [161,161,161,9]
  const float* opacity = (const float*)d_in[3];   // [161,161,161]

  float2* fused = (float2*)d_ws;                  // 161^3 * 8 B ~= 33.4 MB

  const int ncell   = NCELL;
  const int blocks1 = (ncell + FUSE_CELLS_PER_BLOCK - 1) / FUSE_CELLS_PER_BLOCK;
  fuse_kernel<<<blocks1, FUSE_CELLS_PER_BLOCK, 0, stream>>>(grid, opacity, fused, ncell);

  render_kernel<<<NB_RAYS / RENDER_BLOCK, RENDER_BLOCK, 0, stream>>>(
      x, d, fused, (float*)d_out);
}